// LLFullObjectCondensation_43860206027523
// MI455X (gfx1250) — compile-verified
//
#include <hip/hip_runtime.h>
#include <hip/hip_bf16.h>

#define NHITS 80000
#define NTILE 5000          // NHITS / 16
#define NOBJ  512
#define KTILES (NOBJ / 16)  // 32
#define EPSF  1e-9f

// ---------------- workspace layout (byte offsets) ----------------
#define OFF_ALPHA   0        // u64[512]  argmax pack        (4096 B)
#define OFF_CNT     4096     // f32[512]  obj_count
#define OFF_ATT     6144     // f32[512]  sum d2*q over members
#define OFF_REPSUB  8192     // f32[512]  member repulsion (to subtract)
#define OFF_REPALL  10240    // f32[512]  repulsion over ALL hits
#define OFF_QA      12288    // f32[512]  q_alpha
#define OFF_BA      14336    // f32[512]  beta_alpha
#define OFF_NUM     16384    // f32[512]  payload numerator
#define OFF_DEN     18432    // f32[512]  payload denominator
#define OFF_XA      20480    // f32[1024] x_alpha (x,y)
#define OFF_BMAT    24576    // f32[2048] wmma B operand: (-2ax,-2ay,1,|a|^2)
#define OFF_SCAL    32768    // f32[16]   [0]=noise_beta [1]=noise_cnt [2]=cc_sum
#define ZERO_FLOATS 8208     // bytes [0, 32832) zeroed as floats
#define OFF_QARR    32832    // f32[80000] q per hit

typedef float v2f __attribute__((ext_vector_type(2)));
typedef float v8f __attribute__((ext_vector_type(8)));

__device__ __forceinline__ float huber_f(float x, float d) {
    float ax = fabsf(x);
    return ax < d ? x * x : d * d + 2.0f * d * (ax - d);
}
__device__ __forceinline__ float softclip_f(float x, float s) {
    x = x / s;
    x = x > 1.0f ? logf(x + 1.0f) : x;
    return x * s;
}
__device__ __forceinline__ float clip_beta(float b) {
    return fminf(fmaxf(b, 1e-6f), 1.0f - 1e-6f);
}
// raw v_sqrt_f32 (hinge term needs no 0.5-ulp correction sequence)
__device__ __forceinline__ float fast_sqrtf(float x) {
    return __builtin_amdgcn_sqrtf(x);
}

// ---------------- kernel 0: zero accumulators ----------------
__global__ void init_kernel(float* ws) {
    int i = blockIdx.x * blockDim.x + threadIdx.x;
    if (i < ZERO_FLOATS) ws[i] = 0.0f;
}

// ---------------- kernel 1: per-hit pass ----------------
__global__ __launch_bounds__(256)
void pass1_kernel(const float* __restrict__ pred_beta, const float* __restrict__ cc,
                  const float* __restrict__ pred_energy, const float* __restrict__ pred_pos,
                  const float* __restrict__ pred_time, const float* __restrict__ pred_id,
                  const int* __restrict__ t_idx, const float* __restrict__ t_energy,
                  const float* __restrict__ t_pos, const float* __restrict__ t_time,
                  unsigned long long* __restrict__ alphaPack, float* __restrict__ obj_count,
                  float* __restrict__ num_pay, float* __restrict__ den_pay,
                  float* __restrict__ scalars, float* __restrict__ qarr) {
    __shared__ float s_nb, s_nc, s_cc;
    if (threadIdx.x == 0) { s_nb = 0.f; s_nc = 0.f; s_cc = 0.f; }
    __syncthreads();

    int i = blockIdx.x * blockDim.x + threadIdx.x;
    float nb = 0.f, ncnt = 0.f, ccs = 0.f;
    if (i < NHITS) {
        float beta = clip_beta(pred_beta[i]);
        float at = atanhf(beta);
        float q = at * at + 0.1f;           // Q_MIN
        qarr[i] = q;
        float x = cc[2 * i], y = cc[2 * i + 1];
        ccs = x * x + y * y;

        int k = t_idx[i];
        if (k >= 0) {
            atomicAdd(&obj_count[k], 1.0f);
            unsigned long long pack =
                ((unsigned long long)__float_as_uint(beta) << 32) |
                (unsigned long long)(0xFFFFFFFFu - (unsigned)i);   // first-index tiebreak
            atomicMax(&alphaPack[k], pack);

            // payload losses
            float tE = t_energy[i];
            float ediff = fabsf(tE - pred_energy[i]);
            float el = 10.0f * expf(-0.1f * ediff * ediff) + 0.01f * ediff;
            el = softclip_f(el, 10.0f);

            float dx = t_pos[2 * i] - pred_pos[2 * i];
            float dy = t_pos[2 * i + 1] - pred_pos[2 * i + 1];
            float pl = huber_f(sqrtf((dx * dx + dy * dy) * 0.01f + 0.01f), 10.0f);
            pl = softclip_f(pl, 3.0f);

            float tl = softclip_f(huber_f(t_time[i] - pred_time[i], 2.0f), 6.0f);

            float idsq = 0.f;
            #pragma unroll
            for (int c = 0; c < 6; ++c) { float v = pred_id[6 * i + c]; idsq += v * v; }
            float cl = 1e-8f * idsq * (1.0f / 6.0f);

            float payload = el + pl + tl + cl;
            float ew = tE > 10.0f ? 1.0f : (tE - 0.5f) * (1.0f / 9.5f);
            ew = fmaxf(ew, 0.0f);
            float pw = beta * ew;
            atomicAdd(&num_pay[k], payload * pw);
            atomicAdd(&den_pay[k], pw);
        } else {
            nb = beta; ncnt = 1.0f;
        }
    }
    atomicAdd(&s_nb, nb);
    atomicAdd(&s_nc, ncnt);
    atomicAdd(&s_cc, ccs);
    __syncthreads();
    if (threadIdx.x == 0) {
        atomicAdd(&scalars[0], s_nb);
        atomicAdd(&scalars[1], s_nc);
        atomicAdd(&scalars[2], s_cc);
    }
}

// ---------------- kernel 2: per-object condensation point ----------------
__global__ void obj_setup_kernel(const float* __restrict__ pred_beta, const float* __restrict__ cc,
                                 const float* __restrict__ qarr,
                                 const unsigned long long* __restrict__ alphaPack,
                                 const float* __restrict__ obj_count,
                                 float* __restrict__ x_alpha, float* __restrict__ q_alpha,
                                 float* __restrict__ beta_alpha, float* __restrict__ Bmat) {
    int k = blockIdx.x * blockDim.x + threadIdx.x;
    if (k >= NOBJ) return;
    float xa = 0.f, ya = 0.f, qa = 0.f, ba = 0.f;
    if (obj_count[k] > 0.0f) {
        unsigned long long pack = alphaPack[k];
        int i = (int)(0xFFFFFFFFu - (unsigned)(pack & 0xFFFFFFFFull));
        xa = cc[2 * i]; ya = cc[2 * i + 1];
        qa = qarr[i];
        ba = clip_beta(pred_beta[i]);
    }
    x_alpha[2 * k] = xa; x_alpha[2 * k + 1] = ya;
    q_alpha[k] = qa; beta_alpha[k] = ba;
    // B operand row [k-dim]: d2 = x*(-2ax) + y*(-2ay) + |x|^2*1 + 1*|a|^2
    Bmat[4 * k + 0] = -2.0f * xa;
    Bmat[4 * k + 1] = -2.0f * ya;
    Bmat[4 * k + 2] = 1.0f;
    Bmat[4 * k + 3] = xa * xa + ya * ya;
}

// ---------------- kernel 3: attraction + member-repulsion correction ----------------
__global__ __launch_bounds__(256)
void att_kernel(const float* __restrict__ cc, const float* __restrict__ qarr,
                const int* __restrict__ t_idx, const float* __restrict__ x_alpha,
                float* __restrict__ att_sum, float* __restrict__ repSub) {
    int i = blockIdx.x * blockDim.x + threadIdx.x;
    if (i >= NHITS) return;
    int k = t_idx[i];
    if (k < 0) return;
    float dx = cc[2 * i] - x_alpha[2 * k];
    float dy = cc[2 * i + 1] - x_alpha[2 * k + 1];
    float d2 = dx * dx + dy * dy;
    float q = qarr[i];
    atomicAdd(&att_sum[k], d2 * q);
    float r = fmaxf(1.0f - fast_sqrtf(d2 + 1e-6f), 0.0f);  // same formula as rep path
    atomicAdd(&repSub[k], r * q);
}

// ---------------- kernel 4: N x K repulsion via WMMA f32 16x16x4 ----------------
// One 16-hit x 16-object tile: d2 tile from a single WMMA, then hinge epilogue.
__device__ __forceinline__ float rep_tile(int ht, int n, int hi, v2f b,
                                          const float2* __restrict__ cc2,
                                          const float* __restrict__ qarr) {
    int h16 = ht * 16;
    float2 p = cc2[h16 + n];           // hit for A row = lane%16
    // A operand row m: [x, y, |x|^2, 1]; lanes 0-15 hold k-dims {0,1}, 16-31 hold {2,3}
    v2f a;
    a.x = hi ? (p.x * p.x + p.y * p.y) : p.x;
    a.y = hi ? 1.0f : p.y;

    v8f c = {};
    c = __builtin_amdgcn_wmma_f32_16x16x4_f32(
            false, a, false, b, (short)0, c, false, false);

    // C/D layout: lane holds column n; VGPR v -> row (v + 8*hi)
    int rbase = h16 + 8 * hi;
    float acc = 0.0f;
    #pragma unroll
    for (int v = 0; v < 8; ++v) {
        float r = fmaxf(1.0f - fast_sqrtf(c[v] + 1e-6f), 0.0f);
        acc = fmaf(r, qarr[rbase + v], acc);
    }
    return acc;
}

// Block = 256 threads (8 waves). blockIdx.x selects a 16-object column tile;
// B operand loaded once per wave and reused; 2 independent WMMA chains per iter.
__global__ __launch_bounds__(256)
void rep_kernel(const float* __restrict__ cc, const float* __restrict__ qarr,
                const float* __restrict__ Bmat, float* __restrict__ rep_all) {
    __shared__ float colsum[16];
    int tid = threadIdx.x;
    if (tid < 16) colsum[tid] = 0.0f;
    __syncthreads();

    int lane = tid & 31;
    int wave = tid >> 5;
    int n = lane & 15;                 // column within tile (object)
    int hi = (lane >> 4) & 1;          // upper half-wave -> rows 8..15 / k-dims 2,3
    int j = blockIdx.x * 16 + n;       // global object id

    // B operand: VGPR0 = k rows {0,2}, VGPR1 = k rows {1,3} (lanes 0-15 / 16-31)
    v2f b;
    b.x = hi ? Bmat[4 * j + 2] : Bmat[4 * j + 0];
    b.y = hi ? Bmat[4 * j + 3] : Bmat[4 * j + 1];

    const float2* cc2 = (const float2*)cc;
    float acc0 = 0.0f, acc1 = 0.0f;
    int stride = gridDim.y * 8;        // waves across grid.y
    int ht = blockIdx.y * 8 + wave;
    // 2-way unroll: independent WMMA chains back-to-back (wave-uniform bounds)
    for (; ht + stride < NTILE; ht += 2 * stride) {
        acc0 += rep_tile(ht, n, hi, b, cc2, qarr);
        acc1 += rep_tile(ht + stride, n, hi, b, cc2, qarr);
    }
    if (ht < NTILE) acc0 += rep_tile(ht, n, hi, b, cc2, qarr);

    atomicAdd(&colsum[n], acc0 + acc1);   // LDS reduce across 8 waves x 2 half-waves
    __syncthreads();
    if (tid < 16) atomicAdd(&rep_all[blockIdx.x * 16 + tid], colsum[tid]);
}

// ---------------- kernel 5: final per-object combine + scalar ----------------
__global__ __launch_bounds__(512)
void final_kernel(const float* __restrict__ obj_count, const float* __restrict__ q_alpha,
                  const float* __restrict__ beta_alpha, const float* __restrict__ att_sum,
                  const float* __restrict__ repSub, const float* __restrict__ rep_all,
                  const float* __restrict__ num_pay, const float* __restrict__ den_pay,
                  const float* __restrict__ scalars, float* __restrict__ out) {
    __shared__ float s_att[512], s_rep[512], s_beta[512], s_pay[512], s_n[512];
    int k = threadIdx.x;
    float cnt = obj_count[k];
    float attv = 0.f, repv = 0.f, betav = 0.f, payv = 0.f, nv = 0.f;
    if (cnt > 0.0f) {
        float qa = q_alpha[k];
        attv  = qa * att_sum[k] / (cnt + EPSF);
        repv  = qa * (rep_all[k] - repSub[k]) / ((float)NHITS - cnt + EPSF);
        betav = 1.0f - beta_alpha[k];
        payv  = num_pay[k] / (den_pay[k] + EPSF);
        nv    = 1.0f;
    }
    s_att[k] = attv; s_rep[k] = repv; s_beta[k] = betav; s_pay[k] = payv; s_n[k] = nv;
    __syncthreads();
    for (int s = 256; s > 0; s >>= 1) {
        if (k < s) {
            s_att[k] += s_att[k + s]; s_rep[k] += s_rep[k + s];
            s_beta[k] += s_beta[k + s]; s_pay[k] += s_pay[k + s];
            s_n[k] += s_n[k + s];
        }
        __syncthreads();
    }
    if (k == 0) {
        float n_obj = s_n[0] + EPSF;
        float total = (s_att[0] + s_rep[0] + s_beta[0] + s_pay[0]) / n_obj;
        total += scalars[0] / (scalars[1] + EPSF);               // L_noise
        total += 0.001f * scalars[2] / ((float)NHITS * 2.0f);    // L_cc
        out[0] = total;
    }
}

extern "C" void kernel_launch(void* const* d_in, const int* in_sizes, int n_in,
                              void* d_out, int out_size, void* d_ws, size_t ws_size,
                              hipStream_t stream) {
    const float* pred_beta   = (const float*)d_in[0];
    const float* pred_cc     = (const float*)d_in[1];
    const float* pred_energy = (const float*)d_in[2];
    const float* pred_pos    = (const float*)d_in[3];
    const float* pred_time   = (const float*)d_in[4];
    const float* pred_id     = (const float*)d_in[5];
    const int*   t_idx       = (const int*)d_in[6];
    const float* t_energy    = (const float*)d_in[7];
    const float* t_pos       = (const float*)d_in[8];
    const float* t_time      = (const float*)d_in[9];

    char* ws = (char*)d_ws;
    unsigned long long* alphaPack = (unsigned long long*)(ws + OFF_ALPHA);
    float* obj_count = (float*)(ws + OFF_CNT);
    float* att_sum   = (float*)(ws + OFF_ATT);
    float* repSub    = (float*)(ws + OFF_REPSUB);
    float* rep_all   = (float*)(ws + OFF_REPALL);
    float* q_alpha   = (float*)(ws + OFF_QA);
    float* beta_alpha= (float*)(ws + OFF_BA);
    float* num_pay   = (float*)(ws + OFF_NUM);
    float* den_pay   = (float*)(ws + OFF_DEN);
    float* x_alpha   = (float*)(ws + OFF_XA);
    float* Bmat      = (float*)(ws + OFF_BMAT);
    float* scalars   = (float*)(ws + OFF_SCAL);
    float* qarr      = (float*)(ws + OFF_QARR);

    init_kernel<<<(ZERO_FLOATS + 255) / 256, 256, 0, stream>>>((float*)ws);

    pass1_kernel<<<(NHITS + 255) / 256, 256, 0, stream>>>(
        pred_beta, pred_cc, pred_energy, pred_pos, pred_time, pred_id,
        t_idx, t_energy, t_pos, t_time,
        alphaPack, obj_count, num_pay, den_pay, scalars, qarr);

    obj_setup_kernel<<<NOBJ / 256, 256, 0, stream>>>(
        pred_beta, pred_cc, qarr, alphaPack, obj_count,
        x_alpha, q_alpha, beta_alpha, Bmat);

    att_kernel<<<(NHITS + 255) / 256, 256, 0, stream>>>(
        pred_cc, qarr, t_idx, x_alpha, att_sum, repSub);

    dim3 repGrid(KTILES, 25);   // 32 column tiles x 25 row splits; 200 waves cover 5000 tiles
    rep_kernel<<<repGrid, 256, 0, stream>>>(pred_cc, qarr, Bmat, rep_all);

    final_kernel<<<1, 512, 0, stream>>>(
        obj_count, q_alpha, beta_alpha, att_sum, repSub, rep_all,
        num_pay, den_pay, scalars, (float*)d_out);
}